// ResidualGATConv_44555990728951
// MI455X (gfx1250) — compile-verified
//
#include <hip/hip_runtime.h>

#define N_NODES 50000
#define E_EDGES 800000
#define IN_CH   256
#define OUT_CH  64
#define HEADS   4
#define NEG_SLOPE 0.2f
#define ET (E_EDGES + N_NODES)   // edges + self loops

typedef float v2f __attribute__((ext_vector_type(2)));
typedef float v8f __attribute__((ext_vector_type(8)));

__device__ __forceinline__ v8f wmma_f32(v2f a, v2f b, v8f c) {
  return __builtin_amdgcn_wmma_f32_16x16x4_f32(
      /*neg_a=*/false, a, /*neg_b=*/false, b,
      /*c_mod=*/(short)0, c, /*reuse_a=*/false, /*reuse_b=*/false);
}

// ---------------------------------------------------------------------------
// fp32 WMMA GEMM: C[M,Nb] = A[M,256] * B[256,Nb], row-major, K fixed = IN_CH.
// One wave computes a 16x64 C strip (4 accumulators sharing one A fragment).
// Software-pipelined: next k-step's A fragment + 8 B scalars are loaded into
// fresh registers before the 4 WMMAs of the current step, so the scheduler can
// overlap global loads with V_WMMA_F32_16X16X4_F32 instead of a full
// s_wait_loadcnt 0 per WMMA.  waves-per-block must equal Nb/64.
// ---------------------------------------------------------------------------
__global__ void gemm16x64_wmma_f32(const float* __restrict__ A,
                                   const float* __restrict__ B,
                                   float* __restrict__ C, int Nb) {
  const int wave = threadIdx.x >> 5;
  const int lane = threadIdx.x & 31;
  const int row  = lane & 15;   // M (A/C rows) or N (B cols) within tile
  const int hi   = lane >> 4;   // lane half: K pair select / M+8
  const int mt = blockIdx.x;
  const int ng = wave;          // 64-column group

  const float* arow = A + (size_t)(mt * 16 + row) * IN_CH + 2 * hi; // A[row][k+2hi..]
  const float* bb   = B + (size_t)(2 * hi) * Nb + ng * 64 + row;    // B[k+2hi][colgrp]

  v8f acc0 = {}, acc1 = {}, acc2 = {}, acc3 = {};

  // ---- prologue: fragments for k = 0 ----
  v2f a_c = *(const v2f*)(arow);
  float b0 = bb[0],  b1 = bb[Nb];
  float b2 = bb[16], b3 = bb[Nb + 16];
  float b4 = bb[32], b5 = bb[Nb + 32];
  float b6 = bb[48], b7 = bb[Nb + 48];

  #pragma unroll 2
  for (int k = 4; k < IN_CH; k += 4) {
    // load next k-step into fresh registers (overlaps with WMMAs below)
    v2f a_n = *(const v2f*)(arow + k);
    const float* p0 = bb + (size_t)k * Nb;
    const float* p1 = p0 + Nb;
    float n0 = p0[0],  n1 = p1[0];
    float n2 = p0[16], n3 = p1[16];
    float n4 = p0[32], n5 = p1[32];
    float n6 = p0[48], n7 = p1[48];

    v2f b;
    b.x = b0; b.y = b1; acc0 = wmma_f32(a_c, b, acc0);
    b.x = b2; b.y = b3; acc1 = wmma_f32(a_c, b, acc1);
    b.x = b4; b.y = b5; acc2 = wmma_f32(a_c, b, acc2);
    b.x = b6; b.y = b7; acc3 = wmma_f32(a_c, b, acc3);

    a_c = a_n;
    b0 = n0; b1 = n1; b2 = n2; b3 = n3;
    b4 = n4; b5 = n5; b6 = n6; b7 = n7;
  }
  // ---- epilogue: last k-step ----
  {
    v2f b;
    b.x = b0; b.y = b1; acc0 = wmma_f32(a_c, b, acc0);
    b.x = b2; b.y = b3; acc1 = wmma_f32(a_c, b, acc1);
    b.x = b4; b.y = b5; acc2 = wmma_f32(a_c, b, acc2);
    b.x = b6; b.y = b7; acc3 = wmma_f32(a_c, b, acc3);
  }

  float* crow = C + (size_t)(mt * 16 + hi * 8) * Nb + ng * 64 + row; // M = r + 8*hi
  #pragma unroll
  for (int r = 0; r < 8; ++r) {
    float* cr = crow + (size_t)r * Nb;
    cr[0]  = acc0[r];
    cr[16] = acc1[r];
    cr[32] = acc2[r];
    cr[48] = acc3[r];
  }
}

// ---------------------------------------------------------------------------
// Per-node attention logits: a_src[n,h] = <xw[n,h,:], att_src[h,:]>, same for dst.
// ---------------------------------------------------------------------------
__global__ void node_logits(const float* __restrict__ xw,
                            const float* __restrict__ att_s,
                            const float* __restrict__ att_d,
                            float* __restrict__ as_, float* __restrict__ ad_) {
  int idx = blockIdx.x * blockDim.x + threadIdx.x;
  if (idx >= N_NODES * HEADS) return;
  int n = idx >> 2, h = idx & 3;
  const float4* xp = (const float4*)(xw + (size_t)n * (HEADS * OUT_CH) + h * OUT_CH);
  const float4* sp = (const float4*)(att_s + h * OUT_CH);
  const float4* dp = (const float4*)(att_d + h * OUT_CH);
  float ss = 0.f, sd = 0.f;
  #pragma unroll
  for (int i = 0; i < OUT_CH / 4; ++i) {
    float4 x4 = xp[i], s4 = sp[i], d4 = dp[i];
    ss += x4.x * s4.x + x4.y * s4.y + x4.z * s4.z + x4.w * s4.w;
    sd += x4.x * d4.x + x4.y * d4.y + x4.z * d4.z + x4.w * d4.w;
  }
  as_[idx] = ss;
  ad_[idx] = sd;
}

// ---------------------------------------------------------------------------
// Workspace/output init (every call: d_out doubles as head-sum accumulator).
// ---------------------------------------------------------------------------
__global__ void init_ws(unsigned* __restrict__ m_enc, float* __restrict__ denom,
                        float* __restrict__ outacc) {
  int idx = blockIdx.x * blockDim.x + threadIdx.x;
  if (idx < N_NODES * HEADS) { m_enc[idx] = 0u; denom[idx] = 0.f; }
  if (idx < N_NODES * OUT_CH) outacc[idx] = 0.f;
}

__device__ __forceinline__ float lrelu(float x) { return x > 0.f ? x : NEG_SLOPE * x; }
// order-preserving float <-> uint encoding for atomicMax-based segment max
__device__ __forceinline__ unsigned fenc(float f) {
  unsigned u = __float_as_uint(f);
  return (u & 0x80000000u) ? ~u : (u | 0x80000000u);
}
__device__ __forceinline__ float fdec(unsigned u) {
  return (u & 0x80000000u) ? __uint_as_float(u & 0x7FFFFFFFu) : __uint_as_float(~u);
}

__device__ __forceinline__ void edge_sd(const int* __restrict__ ei, int e, int& s, int& d) {
  if (e < E_EDGES) { s = ei[e]; d = ei[E_EDGES + e]; }
  else             { s = e - E_EDGES; d = s; }      // self loop
}

// Pass 1: segment max of leaky-relu logits over destination nodes.
__global__ void edge_max(const int* __restrict__ ei, const float* __restrict__ as_,
                         const float* __restrict__ ad_, unsigned* __restrict__ m_enc) {
  int idx = blockIdx.x * blockDim.x + threadIdx.x;
  if (idx >= ET * HEADS) return;
  int e = idx >> 2, h = idx & 3, s, d;
  edge_sd(ei, e, s, d);
  float alpha = lrelu(as_[s * HEADS + h] + ad_[d * HEADS + h]);
  atomicMax(&m_enc[d * HEADS + h], fenc(alpha));
}

// Pass 2: denom[d,h] = sum_e exp(alpha - m[d,h])
__global__ void edge_denom(const int* __restrict__ ei, const float* __restrict__ as_,
                           const float* __restrict__ ad_,
                           const unsigned* __restrict__ m_enc,
                           float* __restrict__ denom) {
  int idx = blockIdx.x * blockDim.x + threadIdx.x;
  if (idx >= ET * HEADS) return;
  int e = idx >> 2, h = idx & 3, s, d;
  edge_sd(ei, e, s, d);
  float alpha = lrelu(as_[s * HEADS + h] + ad_[d * HEADS + h]);
  float ev = __expf(alpha - fdec(m_enc[d * HEADS + h]));
  atomicAdd(&denom[d * HEADS + h], ev);
}

// Pass 3: one wave per edge. Fold the head-mean into a single [N,64] accumulator:
// acc[d,c] += sum_h a[e,h] * xw[s,h,c]   (one f32 atomic per (edge, channel)).
// The 1KB xw[src] row is prefetched (global_prefetch_b8) before the softmax
// coefficient math so the random gather is in flight during the exp/div work.
__global__ void edge_scatter(const int* __restrict__ ei, const float* __restrict__ as_,
                             const float* __restrict__ ad_,
                             const unsigned* __restrict__ m_enc,
                             const float* __restrict__ denom,
                             const float* __restrict__ xw, float* __restrict__ acc) {
  int e = (int)((blockIdx.x * (size_t)blockDim.x + threadIdx.x) >> 5);
  int lane = threadIdx.x & 31;
  if (e >= ET) return;                       // uniform per wave
  int s, d;
  edge_sd(ei, e, s, d);

  const float* xs = xw + (size_t)s * (HEADS * OUT_CH);
  // 32 lanes x 32B stride spans the full 1KB source row
  __builtin_prefetch(xs + lane * 8, 0, 3);

  float coef[HEADS];
  #pragma unroll
  for (int h = 0; h < HEADS; ++h) {
    float alpha = lrelu(as_[s * HEADS + h] + ad_[d * HEADS + h]);
    coef[h] = __expf(alpha - fdec(m_enc[d * HEADS + h])) / denom[d * HEADS + h];
  }

  float* ao = acc + (size_t)d * OUT_CH;
  #pragma unroll
  for (int cc = 0; cc < 2; ++cc) {
    int c = lane + cc * 32;
    float v = 0.f;
    #pragma unroll
    for (int h = 0; h < HEADS; ++h) v += coef[h] * xs[h * OUT_CH + c];
    atomicAdd(&ao[c], v);
  }
}

// Final: mean over heads + bias + residual + ReLU (in place on d_out).
__global__ void finalize(float* __restrict__ out, const float* __restrict__ xres,
                         const float* __restrict__ bias) {
  int idx = blockIdx.x * blockDim.x + threadIdx.x;
  if (idx >= N_NODES * OUT_CH) return;
  int c = idx & (OUT_CH - 1);
  float v = out[idx] * (1.0f / HEADS) + bias[c] + xres[idx];
  out[idx] = fmaxf(v, 0.f);
}

extern "C" void kernel_launch(void* const* d_in, const int* in_sizes, int n_in,
                              void* d_out, int out_size, void* d_ws, size_t ws_size,
                              hipStream_t stream) {
  const float* x     = (const float*)d_in[0];
  const int*   ei    = (const int*)d_in[1];     // [2,E] int32 (JAX x64 off)
  const float* W     = (const float*)d_in[2];   // [256, 256]
  const float* att_s = (const float*)d_in[3];   // [4, 64]
  const float* att_d = (const float*)d_in[4];   // [4, 64]
  const float* bias  = (const float*)d_in[5];   // [64]
  const float* W_res = (const float*)d_in[6];   // [256, 64]
  float* out = (float*)d_out;

  float*    xw    = (float*)d_ws;                               // N*256
  float*    xres  = xw + (size_t)N_NODES * (HEADS * OUT_CH);    // N*64
  float*    as_   = xres + (size_t)N_NODES * OUT_CH;            // N*4
  float*    ad_   = as_ + (size_t)N_NODES * HEADS;              // N*4
  unsigned* m_enc = (unsigned*)(ad_ + (size_t)N_NODES * HEADS); // N*4
  float*    denom = (float*)(m_enc + (size_t)N_NODES * HEADS);  // N*4

  // GEMMs on the fp32 matrix pipe (V_WMMA_F32_16X16X4_F32).
  // W: 4 waves/block cover all 256 cols; W_res: 1 wave covers 64 cols.
  gemm16x64_wmma_f32<<<dim3(N_NODES / 16, 1), 128, 0, stream>>>(
      x, W, xw, HEADS * OUT_CH);
  gemm16x64_wmma_f32<<<dim3(N_NODES / 16, 1), 32, 0, stream>>>(
      x, W_res, xres, OUT_CH);

  node_logits<<<(N_NODES * HEADS + 255) / 256, 256, 0, stream>>>(xw, att_s, att_d, as_, ad_);
  init_ws<<<(N_NODES * OUT_CH + 255) / 256, 256, 0, stream>>>(m_enc, denom, out);

  edge_max  <<<(ET * HEADS + 255) / 256, 256, 0, stream>>>(ei, as_, ad_, m_enc);
  edge_denom<<<(ET * HEADS + 255) / 256, 256, 0, stream>>>(ei, as_, ad_, m_enc, denom);
  edge_scatter<<<(ET + 7) / 8, 256, 0, stream>>>(ei, as_, ad_, m_enc, denom, xw, out);

  finalize<<<(N_NODES * OUT_CH + 255) / 256, 256, 0, stream>>>(out, xres, bias);
}